// DECOMP_74122545594746
// MI455X (gfx1250) — compile-verified
//
#include <hip/hip_runtime.h>

// EMA scan decomposition for MI455X (gfx1250, wave32).
// x: [B=32, L=4096, C=512] f32.  res = x - ema(x), ema.
// Strategy: blocked scan; 16-timestep tiles computed as S = M*X + d*carry
// via chained V_WMMA_F32_16X16X4_F32 (full f32 precision matrix path).

#define ALPHA 0.3f
#define BETA  0.7f
#define B_    32
#define L_    4096
#define C_    512
#define NCHUNK 8
#define TCH   (L_ / NCHUNK)   // 512 timesteps per chunk
#define NTILE (TCH / 16)      // 32 WMMA tiles per chunk

typedef __attribute__((ext_vector_type(2))) float v2f;
typedef __attribute__((ext_vector_type(8))) float v8f;

// ---------------- Pass 1: chunk-local EMA end value (zero incoming state) ---
// One thread per (b, chunk g in 0..NCHUNK-2, channel). Coalesced over ch.
__global__ void ema_pass1(const float* __restrict__ x, float* __restrict__ locEnd) {
    int tid  = blockIdx.x * blockDim.x + threadIdx.x;
    int ch   = tid % C_;
    int rest = tid / C_;
    int g    = rest % (NCHUNK - 1);
    int b    = rest / (NCHUNK - 1);
    if (b >= B_) return;
    const float* p = x + ((size_t)b * L_ + (size_t)g * TCH) * C_ + ch;
    float s = 0.0f;
    #pragma unroll 8
    for (int i = 0; i < TCH; ++i)
        s = fmaf(BETA, s, ALPHA * p[(size_t)i * C_]);
    locEnd[((size_t)b * NCHUNK + g) * C_ + ch] = s;
}

// ---------------- Pass 2: combine chunk carries (sequential over 8 chunks) --
// Incoming state for chunk 0 is x[b,0,ch] so that y_0 == x_0 (reference init).
__global__ void ema_pass2(const float* __restrict__ x,
                          const float* __restrict__ locEnd,
                          float* __restrict__ carryIn) {
    int tid = blockIdx.x * blockDim.x + threadIdx.x;
    if (tid >= B_ * C_) return;
    int ch = tid % C_;
    int b  = tid / C_;
    float bT = 1.0f;                      // beta^TCH (underflows to ~0 in f32,
    for (int i = 0; i < TCH; ++i)         //  matching the exact math)
        bT *= BETA;
    float c = x[((size_t)b * L_) * C_ + ch];
    carryIn[((size_t)b * NCHUNK + 0) * C_ + ch] = c;
    for (int g = 1; g < NCHUNK; ++g) {
        c = fmaf(bT, c, locEnd[((size_t)b * NCHUNK + (g - 1)) * C_ + ch]);
        carryIn[((size_t)b * NCHUNK + g) * C_ + ch] = c;
    }
}

// ---------------- Pass 3: WMMA tile scan -----------------------------------
// One wave32 per (16-channel group, chunk, batch). For each 16-step tile:
//   acc[t][n] = beta^{t+1} * carry[n]                (seed accumulator)
//   acc     += M(16x16) * X(16x16)  via 4x v_wmma_f32_16x16x4_f32
// Carry hand-off across tiles via wave32 shuffle (row 15 lives in lane m+16,
// accumulator VGPR 7 per the 16x16 f32 C/D layout).
__global__ __launch_bounds__(32) void ema_pass3(const float* __restrict__ x,
                                                const float* __restrict__ carryIn,
                                                float* __restrict__ res,
                                                float* __restrict__ ma) {
    const int lane = threadIdx.x;      // 0..31
    const int m    = lane & 15;
    const int half = lane >> 4;        // 0 -> rows 0..7 / K pair {0,1}; 1 -> rows 8..15 / K pair {2,3}
    const int b    = blockIdx.z;
    const int g    = blockIdx.y;
    const int ch   = blockIdx.x * 16 + m;

    // beta powers 0..16
    float bp[17];
    bp[0] = 1.0f;
    #pragma unroll
    for (int i = 1; i <= 16; ++i) bp[i] = bp[i - 1] * BETA;

    // A-matrix operands: A_j = M[:, 4j..4j+3], M[t,k] = ALPHA*beta^(t-k) (k<=t).
    // 16x4 f32 A layout: lanes 0-15 hold K={4j,4j+1}, lanes 16-31 hold K={4j+2,4j+3}.
    v2f a[4];
    #pragma unroll
    for (int j = 0; j < 4; ++j) {
        int k0 = 4 * j + 2 * half;
        a[j].x = (k0     <= m) ? ALPHA * bp[m - k0]     : 0.0f;
        a[j].y = (k0 + 1 <= m) ? ALPHA * bp[m - k0 - 1] : 0.0f;
    }

    // Carry decay per accumulator VGPR r (global row = r + 8*half): beta^{row+1}
    float dvec[8];
    #pragma unroll
    for (int r = 0; r < 8; ++r) dvec[r] = bp[r + 8 * half + 1];

    float c = carryIn[((size_t)b * NCHUNK + g) * C_ + ch];

    const float* xb = x + (size_t)b * L_ * C_ + ch;

    for (int tt = 0; tt < NTILE; ++tt) {
        const int t0 = g * TCH + tt * 16;
        const float* xt = xb + (size_t)t0 * C_;

        // B-matrix operands (4x16 f32): lanes 0-15 rows {4j,4j+1}, lanes 16-31 rows {4j+2,4j+3}
        v2f bop[4];
        #pragma unroll
        for (int j = 0; j < 4; ++j) {
            int k0 = 4 * j + 2 * half;
            bop[j].x = xt[(size_t)k0 * C_];
            bop[j].y = xt[(size_t)(k0 + 1) * C_];
        }

        union { v8f v; float f[8]; } acc;
        #pragma unroll
        for (int r = 0; r < 8; ++r) acc.f[r] = dvec[r] * c;

        acc.v = __builtin_amdgcn_wmma_f32_16x16x4_f32(false, a[0], false, bop[0], (short)0, acc.v, false, false);
        acc.v = __builtin_amdgcn_wmma_f32_16x16x4_f32(false, a[1], false, bop[1], (short)0, acc.v, false, false);
        acc.v = __builtin_amdgcn_wmma_f32_16x16x4_f32(false, a[2], false, bop[2], (short)0, acc.v, false, false);
        acc.v = __builtin_amdgcn_wmma_f32_16x16x4_f32(false, a[3], false, bop[3], (short)0, acc.v, false, false);

        // New carry: y[t0+15, ch] lives in lane m+16, accumulator VGPR 7.
        c = __shfl(acc.f[7], m + 16, 32);

        // Emit outputs: this lane holds rows 8*half .. 8*half+7 for channel ch.
        #pragma unroll
        for (int r = 0; r < 8; ++r) {
            int row = 8 * half + r;
            size_t idx = ((size_t)b * L_ + t0 + row) * C_ + ch;
            float mav = acc.f[r];
            ma[idx]  = mav;
            res[idx] = xt[(size_t)row * C_] - mav;  // reload hits WGP$/L2
        }
    }
}

// ---------------- Fallback (if scratch too small): scalar row scan ----------
__global__ void ema_fallback(const float* __restrict__ x,
                             float* __restrict__ res, float* __restrict__ ma) {
    int tid = blockIdx.x * blockDim.x + threadIdx.x;
    if (tid >= B_ * C_) return;
    int ch = tid % C_;
    int b  = tid / C_;
    const float* p = x + (size_t)b * L_ * C_ + ch;
    float s = p[0];
    for (int t = 0; t < L_; ++t) {
        float xv = p[(size_t)t * C_];
        s = (t == 0) ? xv : fmaf(BETA, s, ALPHA * xv);
        size_t idx = ((size_t)b * L_ + t) * C_ + ch;
        ma[idx]  = s;
        res[idx] = xv - s;
    }
}

extern "C" void kernel_launch(void* const* d_in, const int* in_sizes, int n_in,
                              void* d_out, int out_size, void* d_ws, size_t ws_size,
                              hipStream_t stream) {
    const float* x  = (const float*)d_in[0];
    float* res = (float*)d_out;                                // output 0
    float* ma  = (float*)d_out + (size_t)B_ * L_ * C_;         // output 1

    const size_t carrySlot = (size_t)B_ * NCHUNK * C_;         // floats
    const size_t wsNeeded  = 2 * carrySlot * sizeof(float);    // 1 MB

    if (ws_size < wsNeeded) {
        int n = B_ * C_;
        ema_fallback<<<(n + 255) / 256, 256, 0, stream>>>(x, res, ma);
        return;
    }

    float* locEnd  = (float*)d_ws;
    float* carryIn = locEnd + carrySlot;

    int n1 = B_ * (NCHUNK - 1) * C_;
    ema_pass1<<<(n1 + 255) / 256, 256, 0, stream>>>(x, locEnd);

    int n2 = B_ * C_;
    ema_pass2<<<(n2 + 255) / 256, 256, 0, stream>>>(x, locEnd, carryIn);

    dim3 grid3(C_ / 16, NCHUNK, B_);
    ema_pass3<<<grid3, 32, 0, stream>>>(x, carryIn, res, ma);
}